// DGIConv_19181323944509
// MI455X (gfx1250) — compile-verified
//
#include <hip/hip_runtime.h>
#include <hip/hip_bf16.h>

#define N_NODES 50000
#define N_EDGES 800000
#define D 128
#define BN_EPS 1e-5f
#define LDS_STRIDE 260   // 260 % 64 == 4 -> 16 rows hit 16 distinct banks

typedef __attribute__((ext_vector_type(2))) float v2f;
typedef __attribute__((ext_vector_type(8))) float v8f;

// ---------------------------------------------------------------- utilities
__global__ void zero_f32_kernel(float* __restrict__ p, int n) {
    int i = blockIdx.x * blockDim.x + threadIdx.x;
    if (i < n) p[i] = 0.0f;
}

__global__ void degree_kernel(const int* __restrict__ dst, float* __restrict__ deg) {
    int e = blockIdx.x * blockDim.x + threadIdx.x;
    if (e < N_EDGES) atomicAdd(&deg[dst[e]], 1.0f);
}

__global__ void invert_deg_kernel(float* __restrict__ deg) {
    int i = blockIdx.x * blockDim.x + threadIdx.x;
    if (i < N_NODES) deg[i] = 1.0f / fmaxf(deg[i], 1.0f);
}

// ------------------------------------------------- edge gather + scatter-add
// One wave32 per edge; each lane moves 4 consecutive floats (128 = 32*4).
__global__ __launch_bounds__(256)
void scatter_kernel(const float* __restrict__ h, const int* __restrict__ src,
                    const int* __restrict__ dst, float* __restrict__ ssum) {
    int gid  = blockIdx.x * blockDim.x + threadIdx.x;
    int e    = gid >> 5;
    int lane = gid & 31;
    if (e >= N_EDGES) return;
    int s = src[e];
    int d = dst[e];
    const float4 v = *reinterpret_cast<const float4*>(h + (size_t)s * D + lane * 4);
    float* o = ssum + (size_t)d * D + lane * 4;
    atomicAdd(o + 0, v.x);
    atomicAdd(o + 1, v.y);
    atomicAdd(o + 2, v.z);
    atomicAdd(o + 3, v.w);
}

// ------------------------------------------------------ fused SAGE GEMM (WMMA)
// out[n,:] = h[n,:] @ Wself + (ssum[n,:]*inv[n]) @ Wneigh + bias
// Viewed as [h | mean] (N x 256) @ [Wself ; Wneigh] (256 x 128).
// Block = 16 rows (N_NODES = 3125*16 exactly, no tail -> EXEC all ones).
// 8 waves/block, wave w owns output cols [16w, 16w+16).
__global__ __launch_bounds__(256)
void sage_gemm_kernel(const float* __restrict__ h, const float* __restrict__ ssum,
                      const float* __restrict__ inv_deg,
                      const float* __restrict__ Wself, const float* __restrict__ Wneigh,
                      const float* __restrict__ bias, float* __restrict__ out) {
    __shared__ float At[16 * LDS_STRIDE];

    const int row0 = blockIdx.x * 16;

    // Cooperative stage of the 16x256 A-tile ([h | mean]) into LDS.
    for (int i = threadIdx.x; i < 16 * 256; i += 256) {
        int r = i >> 8;
        int c = i & 255;
        float val;
        if (c < D) val = h[(size_t)(row0 + r) * D + c];
        else       val = ssum[(size_t)(row0 + r) * D + (c - D)] * inv_deg[row0 + r];
        At[r * LDS_STRIDE + c] = val;
    }
    __syncthreads();

    const int wave = threadIdx.x >> 5;
    const int lane = threadIdx.x & 31;
    const int half = lane >> 4;    // 0: lanes 0-15 (K=0,1), 1: lanes 16-31 (K=2,3)
    const int l15  = lane & 15;
    const int col0 = wave * 16;

    v8f acc = {0.f, 0.f, 0.f, 0.f, 0.f, 0.f, 0.f, 0.f};

    #pragma unroll 4
    for (int k0 = 0; k0 < 256; k0 += 4) {
        const int ka = k0 + 2 * half;           // even, so ka & ka+1 stay in one half
        v2f a, b;
        // A fragment (16x4 f32): VGPR0 = K={0|2}, VGPR1 = K={1|3}, M = lane&15
        a.x = At[l15 * LDS_STRIDE + ka];
        a.y = At[l15 * LDS_STRIDE + ka + 1];
        // B fragment (4x16 f32): rows striped across lanes, N = lane&15
        const float* __restrict__ Wb = (ka < D) ? Wself : Wneigh;
        const int kk = (ka < D) ? ka : (ka - D);
        b.x = Wb[(size_t)kk       * D + col0 + l15];
        b.y = Wb[(size_t)(kk + 1) * D + col0 + l15];
        acc = __builtin_amdgcn_wmma_f32_16x16x4_f32(
            /*neg_a=*/false, a, /*neg_b=*/false, b,
            /*c_mod=*/(short)0, acc, /*reuse_a=*/false, /*reuse_b=*/false);
    }

    // C/D layout: VGPR p, lanes 0-15 -> M=p, lanes 16-31 -> M=p+8; N = lane&15.
    const float bj = bias[col0 + l15];
    #pragma unroll
    for (int p = 0; p < 8; ++p) {
        out[(size_t)(row0 + p + 8 * half) * D + col0 + l15] = acc[p] + bj;
    }
}

// ---------------------------------------------------------------- batch-norm
__global__ __launch_bounds__(128)
void bn_stats_kernel(const float* __restrict__ h, float* __restrict__ sums,
                     float* __restrict__ sqs) {
    const int col = threadIdx.x;   // one column per thread
    const int rows_per_block = (N_NODES + gridDim.x - 1) / gridDim.x;
    int r0 = blockIdx.x * rows_per_block;
    int r1 = r0 + rows_per_block;
    if (r1 > N_NODES) r1 = N_NODES;
    float s = 0.0f, q = 0.0f;
    for (int r = r0; r < r1; ++r) {
        float v = h[(size_t)r * D + col];
        s += v;
        q += v * v;
    }
    atomicAdd(&sums[col], s);
    atomicAdd(&sqs[col], q);
}

__global__ __launch_bounds__(256)
void bn_apply_relu_kernel(float* __restrict__ h, const float* __restrict__ gamma,
                          const float* __restrict__ beta,
                          const float* __restrict__ sums, const float* __restrict__ sqs) {
    int i = blockIdx.x * blockDim.x + threadIdx.x;
    if (i >= N_NODES * D) return;
    int col = i & (D - 1);
    const float invN = 1.0f / (float)N_NODES;
    float m   = sums[col] * invN;
    float var = sqs[col] * invN - m * m;   // biased variance, matches reference
    float hn  = (h[i] - m) * rsqrtf(var + BN_EPS) * gamma[col] + beta[col];
    h[i] = fmaxf(hn, 0.0f);
}

// ---------------------------------------------------------------------------
extern "C" void kernel_launch(void* const* d_in, const int* in_sizes, int n_in,
                              void* d_out, int out_size, void* d_ws, size_t ws_size,
                              hipStream_t stream) {
    const float* x       = (const float*)d_in[0];
    const int*   src     = (const int*)  d_in[1];
    const int*   dst     = (const int*)  d_in[2];
    const float* Wself0  = (const float*)d_in[3];
    const float* Wneigh0 = (const float*)d_in[4];
    const float* b0      = (const float*)d_in[5];
    const float* Wself1  = (const float*)d_in[6];
    const float* Wneigh1 = (const float*)d_in[7];
    const float* b1      = (const float*)d_in[8];
    const float* Wself2  = (const float*)d_in[9];
    const float* Wneigh2 = (const float*)d_in[10];
    const float* b2      = (const float*)d_in[11];
    const float* gamma0  = (const float*)d_in[12];
    const float* beta0   = (const float*)d_in[13];
    const float* gamma1  = (const float*)d_in[14];
    const float* beta1   = (const float*)d_in[15];
    float* out = (float*)d_out;

    // Workspace layout (all float): bufA | bufB | inv_deg | sums | sqs
    float* bufA = (float*)d_ws;                  // N*D
    float* bufB = bufA + (size_t)N_NODES * D;    // N*D
    float* inv  = bufB + (size_t)N_NODES * D;    // N
    float* sums = inv + N_NODES;                 // 128
    float* sqs  = sums + D;                      // 128

    const int ND        = N_NODES * D;
    const int zeroGrid  = (ND + 255) / 256;
    const int nodeGrid  = (N_NODES + 255) / 256;
    const int edgeGrid  = (N_EDGES + 255) / 256;
    const int scatGrid  = (N_EDGES * 32 + 255) / 256;  // wave per edge
    const int gemmGrid  = N_NODES / 16;                 // 3125, exact
    const int bnGrid    = 200;

    // Degree (layer-invariant): compute once, invert in place.
    zero_f32_kernel<<<nodeGrid, 256, 0, stream>>>(inv, N_NODES);
    degree_kernel<<<edgeGrid, 256, 0, stream>>>(dst, inv);
    invert_deg_kernel<<<nodeGrid, 256, 0, stream>>>(inv);

    // ---- Layer 0 : x -> bufA
    zero_f32_kernel<<<zeroGrid, 256, 0, stream>>>(bufA, ND);
    scatter_kernel<<<scatGrid, 256, 0, stream>>>(x, src, dst, bufA);
    sage_gemm_kernel<<<gemmGrid, 256, 0, stream>>>(x, bufA, inv, Wself0, Wneigh0, b0, bufA);
    zero_f32_kernel<<<1, 256, 0, stream>>>(sums, 2 * D);
    bn_stats_kernel<<<bnGrid, 128, 0, stream>>>(bufA, sums, sqs);
    bn_apply_relu_kernel<<<zeroGrid, 256, 0, stream>>>(bufA, gamma0, beta0, sums, sqs);

    // ---- Layer 1 : bufA -> bufB
    zero_f32_kernel<<<zeroGrid, 256, 0, stream>>>(bufB, ND);
    scatter_kernel<<<scatGrid, 256, 0, stream>>>(bufA, src, dst, bufB);
    sage_gemm_kernel<<<gemmGrid, 256, 0, stream>>>(bufA, bufB, inv, Wself1, Wneigh1, b1, bufB);
    zero_f32_kernel<<<1, 256, 0, stream>>>(sums, 2 * D);
    bn_stats_kernel<<<bnGrid, 128, 0, stream>>>(bufB, sums, sqs);
    bn_apply_relu_kernel<<<zeroGrid, 256, 0, stream>>>(bufB, gamma1, beta1, sums, sqs);

    // ---- Layer 2 : bufB -> out
    zero_f32_kernel<<<zeroGrid, 256, 0, stream>>>(bufA, ND);
    scatter_kernel<<<scatGrid, 256, 0, stream>>>(bufB, src, dst, bufA);
    sage_gemm_kernel<<<gemmGrid, 256, 0, stream>>>(bufB, bufA, inv, Wself2, Wneigh2, b2, out);
}